// AttnResBlock_9655086481588
// MI455X (gfx1250) — compile-verified
//
#include <hip/hip_runtime.h>
#include <math.h>

#ifndef __has_builtin
#define __has_builtin(x) 0
#endif

#define N_DEPTH 13
#define B_DIM 2
#define T_DIM 2048
#define D_DIM 1024
#define W_ROW 13                 // LDS row 13 holds rms_weight * w_proj
#define ROWSTRIDE 1028           // floats; +4 pad -> conflict-free WMMA A reads

typedef float v2f __attribute__((ext_vector_type(2)));
typedef float v4f __attribute__((ext_vector_type(4)));
typedef float v8f __attribute__((ext_vector_type(8)));
typedef int   v4i __attribute__((vector_size(4 * sizeof(int))));

#define AS1 __attribute__((address_space(1)))
#define AS3 __attribute__((address_space(3)))

#define HAVE_WMMA_F32X4 __has_builtin(__builtin_amdgcn_wmma_f32_16x16x4_f32)
#define HAVE_ASYNC_LDS  __has_builtin(__builtin_amdgcn_global_load_async_to_lds_b128)

__global__ __launch_bounds__(256, 1) void attn_res_fused(
    const float* __restrict__ V,
    const float* __restrict__ rms_w,
    const float* __restrict__ w_proj,
    float* __restrict__ out)
{
    __shared__ float rows[14 * ROWSTRIDE];   // 57,568 B: 13 V rows + w_comb row
    __shared__ float gram[256];              // 16x16 Gram accumulator
    __shared__ float alpha_s[16];

    const int tid = threadIdx.x;             // 256 threads = 8 wave32
    const int bt  = blockIdx.x;              // one (b,t) per workgroup
    const int b   = bt >> 11;                // / 2048
    const int t   = bt & 2047;

    // ---- stage row 13: combined weight w_comb = rms_weight * w_proj ----
    {
        const int d = tid << 2;
        v4f rw = *(const v4f*)(rms_w + d);
        v4f wp = *(const v4f*)(w_proj + d);
        *(v4f*)&rows[W_ROW * ROWSTRIDE + d] = rw * wp;
    }

    // ---- stage the 13 depth rows of V for this (b,t) into LDS (read V ONCE) ----
    const size_t baseBT = ((size_t)b * T_DIM + t) * (size_t)D_DIM;
    const size_t rowG   = (size_t)B_DIM * T_DIM * D_DIM;   // stride between depth slabs
    #pragma unroll
    for (int n = 0; n < N_DEPTH; ++n) {
        const float* g = V + baseBT + (size_t)n * rowG + (size_t)(tid << 2);
#if HAVE_ASYNC_LDS
        __builtin_amdgcn_global_load_async_to_lds_b128(
            (AS1 v4i*)(v4i*)(void*)g,
            (AS3 v4i*)(v4i*)(void*)&rows[n * ROWSTRIDE + (tid << 2)],
            0, 0);
#else
        *(v4f*)&rows[n * ROWSTRIDE + (tid << 2)] =
            __builtin_nontemporal_load((const v4f*)g);
#endif
    }

    gram[tid] = 0.0f;

#if HAVE_ASYNC_LDS
#if __has_builtin(__builtin_amdgcn_s_wait_asynccnt)
    __builtin_amdgcn_s_wait_asynccnt(0);
#else
    asm volatile("s_wait_asynccnt 0" ::: "memory");
#endif
#endif
    __syncthreads();

#if HAVE_WMMA_F32X4
    // ---- Gram matrix via WMMA: acc[m,m'] = sum_d row_m[d] * row_m'[d].
    // 16x4 fp32 A layout == transposed B layout, so wmma(A, A, acc) = A*A^T.
    // Diagonal -> sum of squares (RMS), column 13 -> dot with w_comb (logits).
    {
        const int lane = tid & 31;
        const int wave = tid >> 5;                 // each wave owns 128 d-columns
        int rrow = lane & 15;
        if (rrow > W_ROW) rrow = W_ROW;            // pad rows 14/15: duplicate w row (unused)
        const float* ap = &rows[rrow * ROWSTRIDE + (wave << 7) + ((lane >> 4) << 1)];
        v8f acc = {0.f, 0.f, 0.f, 0.f, 0.f, 0.f, 0.f, 0.f};
        #pragma unroll
        for (int s = 0; s < 32; ++s) {
            v2f a = *(const v2f*)(ap + (s << 2));
            acc = __builtin_amdgcn_wmma_f32_16x16x4_f32(
                false, a, false, a, (short)0, acc, false, false);
        }
        // deterministic cross-wave reduction (each wave covers all 256 Gram slots)
        const int Mbase = (lane >> 4) << 3;
        const int Nn    = lane & 15;
        for (int w = 0; w < 8; ++w) {
            if (wave == w) {
                #pragma unroll
                for (int r = 0; r < 8; ++r)
                    gram[(Mbase + r) * 16 + Nn] += acc[r];
            }
            __syncthreads();
        }
    }
#else
    // Fallback (no f32 WMMA builtin): deterministic scalar dot products.
    if (tid < 2 * N_DEPTH) {
        const int n      = tid >> 1;
        const int do_dot = tid & 1;
        const float* r = &rows[n * ROWSTRIDE];
        const float* w = &rows[W_ROW * ROWSTRIDE];
        float s = 0.f;
        for (int d = 0; d < D_DIM; ++d)
            s = fmaf(r[d], do_dot ? w[d] : r[d], s);
        gram[n * 16 + (do_dot ? W_ROW : n)] = s;
    }
    __syncthreads();
#endif

    // ---- logits + softmax over the 13 depth entries (tiny; one lane) ----
    if (tid == 0) {
        float lg[N_DEPTH];
        float mx = -3.0e38f;
        #pragma unroll
        for (int n = 0; n < N_DEPTH; ++n) {
            const float ss = gram[n * 16 + n];          // sum_d V^2
            const float dp = gram[n * 16 + W_ROW];      // sum_d V * w_comb
            const float l  = rsqrtf(ss * (1.0f / D_DIM) + 1.1920929e-07f) * dp;
            lg[n] = l;
            mx = fmaxf(mx, l);
        }
        float sum = 0.f;
        #pragma unroll
        for (int n = 0; n < N_DEPTH; ++n) {
            const float e = __expf(lg[n] - mx);
            lg[n] = e;
            sum += e;
        }
        const float inv = 1.0f / sum;
        #pragma unroll
        for (int n = 0; n < N_DEPTH; ++n)
            alpha_s[n] = lg[n] * inv;
    }
    __syncthreads();

    // ---- weighted sum over depth of UN-normalized V (from LDS), coalesced store ----
    {
        const int d = tid << 2;
        v4f h = {0.f, 0.f, 0.f, 0.f};
        #pragma unroll
        for (int n = 0; n < N_DEPTH; ++n) {
            const float an = alpha_s[n];
            v4f v = *(const v4f*)&rows[n * ROWSTRIDE + d];
            h += an * v;
        }
        *(v4f*)(out + baseBT + d) = h;
    }
}

extern "C" void kernel_launch(void* const* d_in, const int* in_sizes, int n_in,
                              void* d_out, int out_size, void* d_ws, size_t ws_size,
                              hipStream_t stream) {
    (void)in_sizes; (void)n_in; (void)d_ws; (void)ws_size; (void)out_size;
    const float* V     = (const float*)d_in[0];
    const float* rms_w = (const float*)d_in[1];
    const float* wproj = (const float*)d_in[2];
    float* out = (float*)d_out;

    dim3 grid(B_DIM * T_DIM);   // 4096 workgroups, one per (b,t)
    dim3 block(256);            // 8 wave32
    attn_res_fused<<<grid, block, 0, stream>>>(V, rms_w, wproj, out);
}